// JK_CFLP_60988535603568
// MI455X (gfx1250) — compile-verified
//
#include <hip/hip_runtime.h>
#include <hip/hip_bf16.h>

typedef __attribute__((ext_vector_type(2))) float v2f;
typedef __attribute__((ext_vector_type(4))) float v4f;
typedef __attribute__((ext_vector_type(8))) float v8f;

#define NFEAT_OUT 64  // hidden/class width (all layer outputs are 64 wide)

// ---------------------------------------------------------------- utilities
__global__ void zero_f32_kernel(float* __restrict__ p, long long n) {
  long long t = (long long)blockIdx.x * blockDim.x + threadIdx.x;
  if (t < n) p[t] = 0.0f;
}

// Count degrees with float atomics (matches reference segment_sum of ones).
__global__ void degree_kernel(const int* __restrict__ src,
                              const int* __restrict__ dst,
                              float* __restrict__ deg_out,
                              float* __restrict__ deg_in, int E) {
  int e = blockIdx.x * blockDim.x + threadIdx.x;
  if (e >= E) return;
  atomicAdd(&deg_out[src[e]], 1.0f);
  atomicAdd(&deg_in[dst[e]], 1.0f);
}

__global__ void norm_kernel(const float* __restrict__ deg_out,
                            const float* __restrict__ deg_in,
                            float* __restrict__ norm_out,
                            float* __restrict__ norm_in, int n) {
  int i = blockIdx.x * blockDim.x + threadIdx.x;
  if (i >= n) return;
  float o = deg_out[i];
  float d = deg_in[i];
  norm_out[i] = (o > 0.0f) ? rsqrtf(fmaxf(o, 1.0f)) : 0.0f;
  norm_in[i]  = (d > 0.0f) ? rsqrtf(fmaxf(d, 1.0f)) : 0.0f;
}

// ------------------------------------------------------ edge gather/scatter
// One thread per (edge, 4-feature chunk): 16 threads cover an edge's 64
// features. Gather is one b128 load; scatter is 4 f32 atomics (irreducible).
__global__ void scatter_kernel(const float* __restrict__ X,
                               const int* __restrict__ src,
                               const int* __restrict__ dst,
                               const float* __restrict__ norm_src,
                               float* __restrict__ agg, long long E) {
  long long t = (long long)blockIdx.x * blockDim.x + threadIdx.x;
  long long e = t >> 4;          // 16 chunks of 4 feats per edge
  int f4 = (int)(t & 15);
  if (e >= E) return;
  int s = src[e];
  int d = dst[e];
  float ns = norm_src[s];
  const v4f x = *(const v4f*)(X + (size_t)s * NFEAT_OUT + f4 * 4);
  float* ap = agg + (size_t)d * NFEAT_OUT + f4 * 4;
  atomicAdd(ap + 0, x.x * ns);
  atomicAdd(ap + 1, x.y * ns);
  atomicAdd(ap + 2, x.z * ns);
  atomicAdd(ap + 3, x.w * ns);
}

// Layer-1 epilogue: h = relu(agg * norm_in + b); also initializes JK-max out.
// One thread per (node, 4-feature chunk): b128 in, 2x b128 out.
__global__ void h1_epilogue_kernel(const float* __restrict__ agg,
                                   const float* __restrict__ norm_in,
                                   const float* __restrict__ bias,
                                   float* __restrict__ h,
                                   float* __restrict__ out, int M) {
  long long t = (long long)blockIdx.x * blockDim.x + threadIdx.x;
  int i = (int)(t >> 4);
  int f4 = (int)(t & 15);
  if (i >= M) return;
  float ni = norm_in[i];
  const v4f a = *(const v4f*)(agg + t * 4);
  const v4f b = *(const v4f*)(bias + f4 * 4);
  v4f v;
  v.x = fmaxf(a.x * ni + b.x, 0.0f);
  v.y = fmaxf(a.y * ni + b.y, 0.0f);
  v.z = fmaxf(a.z * ni + b.z, 0.0f);
  v.w = fmaxf(a.w * ni + b.w, 0.0f);
  *(v4f*)(h + t * 4) = v;
  *(v4f*)(out + t * 4) = v;   // initialize running JK max
}

// -------------------------------------------------------------- WMMA GEMM
// H[M,N] = op( (A * rowscale) @ W + bias ), op = relu if relu!=0,
// and optionally maxout = max(maxout, H) (JK-max accumulate).
// One wave computes one 16x16 tile via V_WMMA_F32_16X16X4_F32; 4 waves per
// block cover N=64; grid.x covers M/16 row tiles. K in {128, 64}.
__global__ void gemm_wmma_f32_kernel(const float* __restrict__ A,
                                     const float* __restrict__ rowscale,
                                     const float* __restrict__ W,
                                     const float* __restrict__ bias,
                                     float* __restrict__ H,
                                     float* __restrict__ maxout,
                                     int M, int K, int N, int relu) {
  const int wave = threadIdx.x >> 5;   // 0..3 -> N tile
  const int lane = threadIdx.x & 31;
  const int lr   = lane & 15;
  const int hi   = lane >> 4;          // 0: K=k,k+1 ; 1: K=k+2,k+3
  const int row0 = blockIdx.x * 16;
  const int col0 = wave * 16;

  const int arow = row0 + lr;          // A-fragment row for this lane
  const float scale = (rowscale != nullptr) ? rowscale[arow] : 1.0f;
  const float* __restrict__ ap = A + (size_t)arow * K;

  v8f acc = {};
  for (int k = 0; k < K; k += 4) {
    const int ka = k + 2 * hi;
    v2f a, b;
    a.x = ap[ka]     * scale;          // A[arow][ka]
    a.y = ap[ka + 1] * scale;          // A[arow][ka+1]
    b.x = W[(size_t)ka       * N + col0 + lr];  // B[ka][col]
    b.y = W[(size_t)(ka + 1) * N + col0 + lr];  // B[ka+1][col]
    acc = __builtin_amdgcn_wmma_f32_16x16x4_f32(
        /*neg_a=*/false, a, /*neg_b=*/false, b,
        /*c_mod=*/(short)0, acc, /*reuse_a=*/false, /*reuse_b=*/false);
  }

  // C/D layout: VGPR v, lanes 0-15 -> M=v, lanes 16-31 -> M=v+8; N = lr.
  const int c = col0 + lr;
#pragma unroll
  for (int v = 0; v < 8; ++v) {
    const int r = row0 + v + 8 * hi;
    if (r >= M) continue;
    float val = acc[v];
    if (bias) val += bias[c];
    if (relu) val = fmaxf(val, 0.0f);
    H[(size_t)r * N + c] = val;
    if (maxout) {
      float* p = &maxout[(size_t)r * N + c];
      *p = fmaxf(*p, val);   // each (r,c) owned by exactly one thread
    }
  }
}

// ------------------------------------------------------------------ driver
extern "C" void kernel_launch(void* const* d_in, const int* in_sizes, int n_in,
                              void* d_out, int out_size, void* d_ws, size_t ws_size,
                              hipStream_t stream) {
  const float* features = (const float*)d_in[0];   // [N,128]
  const int*   src      = (const int*)  d_in[1];   // [E]
  const int*   dst      = (const int*)  d_in[2];   // [E]
  const float* W1 = (const float*)d_in[3];         // [128,64]
  const float* b1 = (const float*)d_in[4];         // [64]
  const float* W2 = (const float*)d_in[5];         // [64,64]
  const float* b2 = (const float*)d_in[6];         // [64]
  const float* W3 = (const float*)d_in[7];         // [64,64]
  const float* b3 = (const float*)d_in[8];         // [64]

  const int IN_FEATS = 128;
  const int Nn = in_sizes[0] / IN_FEATS;           // 50000 nodes
  const long long E = in_sizes[1];                 // 850000 edges
  float* out = (float*)d_out;                      // [Nn,64]

  // workspace carve-up (floats)
  float* ws = (float*)d_ws;
  float* deg_out  = ws;                 // Nn
  float* deg_in   = deg_out + Nn;       // Nn
  float* norm_out = deg_in + Nn;        // Nn
  float* norm_in  = norm_out + Nn;      // Nn
  float* xw  = norm_in + Nn;            // Nn*64  (GEMM output / layer input)
  float* agg = xw + (size_t)Nn * 64;    // Nn*64  (edge aggregation)
  float* h   = agg + (size_t)Nn * 64;   // Nn*64  (current layer output)

  const int TPB = 256;
  const long long nf  = (long long)Nn * 64;  // node-feature elements
  const long long nc  = (long long)Nn * 16;  // node float4 chunks
  const long long ec  = E * 16;              // edge float4 chunks
  const int mtiles = (Nn + 15) / 16;

  // 1) degrees + norms
  zero_f32_kernel<<<(int)((2LL * Nn + TPB - 1) / TPB), TPB, 0, stream>>>(deg_out, 2LL * Nn);
  degree_kernel<<<(int)((E + TPB - 1) / TPB), TPB, 0, stream>>>(src, dst, deg_out, deg_in, (int)E);
  norm_kernel<<<(Nn + TPB - 1) / TPB, TPB, 0, stream>>>(deg_out, deg_in, norm_out, norm_in, Nn);

  // 2) layer 1: xw = features @ W1 (no epilogue; W1 shrinks dims -> GEMM first)
  gemm_wmma_f32_kernel<<<mtiles, 128, 0, stream>>>(
      features, nullptr, W1, nullptr, xw, nullptr, Nn, IN_FEATS, 64, 0);
  zero_f32_kernel<<<(int)((nf + TPB - 1) / TPB), TPB, 0, stream>>>(agg, nf);
  scatter_kernel<<<(int)((ec + TPB - 1) / TPB), TPB, 0, stream>>>(xw, src, dst, norm_out, agg, E);
  h1_epilogue_kernel<<<(int)((nc + TPB - 1) / TPB), TPB, 0, stream>>>(agg, norm_in, b1, h, out, Nn);

  // 3) layer 2: agg = scatter(h); h = relu((agg*norm_in)@W2 + b2); out=max(out,h)
  zero_f32_kernel<<<(int)((nf + TPB - 1) / TPB), TPB, 0, stream>>>(agg, nf);
  scatter_kernel<<<(int)((ec + TPB - 1) / TPB), TPB, 0, stream>>>(h, src, dst, norm_out, agg, E);
  gemm_wmma_f32_kernel<<<mtiles, 128, 0, stream>>>(
      agg, norm_in, W2, b2, h, out, Nn, 64, 64, 1);

  // 4) layer 3: same with W3
  zero_f32_kernel<<<(int)((nf + TPB - 1) / TPB), TPB, 0, stream>>>(agg, nf);
  scatter_kernel<<<(int)((ec + TPB - 1) / TPB), TPB, 0, stream>>>(h, src, dst, norm_out, agg, E);
  gemm_wmma_f32_kernel<<<mtiles, 128, 0, stream>>>(
      agg, norm_in, W3, b3, h, out, Nn, 64, 64, 1);
}